// CRFLayer_8916352106579
// MI455X (gfx1250) — compile-verified
//
#include <hip/hip_runtime.h>
#include <hip/hip_bf16.h>

// CRF forward (log-partition) on gfx1250, log2-domain, latency-optimized.
// One wave32 per 16-batch tile (128 single-wave workgroups).  Per step:
//   beta'[b,j] = feat[b,j]*L2E + m2[b] + log2( sum_i exp2(beta[b,i]-m2[b]) * exp2(T[j,i]*L2E) )
// exp2-domain GEMM = 4 independent 4-deep V_WMMA_F32_16X16X4_F32 chains.
// All sync is intra-wave: LDS is hardware in-order per wave, so only a compiler
// scheduling fence (__builtin_amdgcn_wave_barrier) is needed -- no loadcnt flush,
// feat loads overlap the entire WMMA section.  Mask blend is branchless fma
// (mask is {0,1}; matches reference's new*m + alpha*(1-m)).

typedef __attribute__((ext_vector_type(2))) float v2f;
typedef __attribute__((ext_vector_type(8))) float v8f;

#define NCLS 32
#define SOS_ 30
#define EOS_ 31
#define NEGV (-10000.0f)
#define L2E  1.44269504088896340736f   // log2(e)
#define LN2  0.69314718055994530942f

__global__ __launch_bounds__(32)
void crf_fwd_wmma(const float* __restrict__ feats,
                  const float* __restrict__ mask,
                  const float* __restrict__ trans,
                  float* __restrict__ out,
                  int seq_len, int batch)
{
    __shared__ float tS[NCLS * NCLS];   // transition staged in LDS
    __shared__ float vS[16 * 33];       // exp2(beta - m2), padded rows

    const int lane = threadIdx.x & 31;  // wave32
    const int hi   = lane >> 4;
    const int ln   = lane & 15;
    const int b0   = blockIdx.x * 16;   // batch tile base

    // ---- stage T (32x32 f32 = 4KB) into LDS with b128 loads ----
    {
        const float4* g4 = (const float4*)trans;
        float4* s4 = (float4*)tS;
        #pragma unroll
        for (int q = 0; q < 8; ++q)
            s4[q * 32 + lane] = g4[q * 32 + lane];
    }
    __syncthreads();   // once, outside the hot loop

    // ---- B operand registers: B[K=i, N=j] = exp2(T[j,i]*L2E) ----
    // f32 4x16 B layout: VGPR p, lane -> (K = 2*hi + p, N = ln); k-step kk adds 4*kk.
    v2f Bk[2][8];
    #pragma unroll
    for (int tau = 0; tau < 2; ++tau) {
        #pragma unroll
        for (int kk = 0; kk < 8; ++kk) {
            const int j  = 16 * tau + ln;
            const int i0 = 4 * kk + 2 * hi;
            Bk[tau][kk].x = __builtin_amdgcn_exp2f(tS[j * NCLS + i0]     * L2E);
            Bk[tau][kk].y = __builtin_amdgcn_exp2f(tS[j * NCLS + i0 + 1] * L2E);
        }
    }
    const float tEOS0 = tS[EOS_ * NCLS + ln]      * L2E;  // T[EOS, ln]   (log2 domain)
    const float tEOS1 = tS[EOS_ * NCLS + 16 + ln] * L2E;  // T[EOS, ln+16]

    // ---- beta in C/D register layout: reg r, lane -> (b = r + 8*hi, j = ln [+16]) ----
    float a0[8], a1[8];
    #pragma unroll
    for (int r = 0; r < 8; ++r) {
        a0[r] = NEGV * L2E;                                  // j in [0,16): never SOS(30)
        a1[r] = (ln == (SOS_ - 16)) ? 0.0f : NEGV * L2E;     // j = 16+ln == 30
    }

    float mD[8];   // per-b scale offset (refreshed on even steps; uniform per half)

    for (int t = 0; t < seq_len; ++t) {
        const float* ft = feats + (size_t)t * batch * NCLS + (size_t)b0 * NCLS;
        const float* mt = mask  + (size_t)t * batch + b0;

        // prefetch next timestep's 2KB feat tile (global_prefetch_b8)
        if (t + 1 < seq_len) {
            const float* fn = ft + (size_t)batch * NCLS;
            #pragma unroll
            for (int q = 0; q < 4; ++q)
                __builtin_prefetch(fn + q * 128 + lane * 4, 0, 3);
        }

        // feat/mask loads issued early; consumed only in postprocess (no fence between)
        float f0[8], f1[8], mk[8];
        #pragma unroll
        for (int r = 0; r < 8; ++r) {
            const int b = r + 8 * hi;
            f0[r] = ft[b * NCLS + ln];
            f1[r] = ft[b * NCLS + 16 + ln];
            mk[r] = mt[b];
        }

        // refresh per-b max every 2nd step (butterfly within each 16-lane half);
        // odd steps reuse the stale offset: still uniform per b -> algebra exact.
        if ((t & 1) == 0) {
            #pragma unroll
            for (int r = 0; r < 8; ++r) {
                float v = fmaxf(a0[r], a1[r]);
                v = fmaxf(v, __shfl_xor(v, 1, 32));
                v = fmaxf(v, __shfl_xor(v, 2, 32));
                v = fmaxf(v, __shfl_xor(v, 4, 32));
                v = fmaxf(v, __shfl_xor(v, 8, 32));
                mD[r] = v;
            }
        }

        // v = exp2(beta - m2) staged to LDS in row-major [b][i]
        #pragma unroll
        for (int r = 0; r < 8; ++r) {
            const int b = r + 8 * hi;
            vS[b * 33 + ln]      = __builtin_amdgcn_exp2f(a0[r] - mD[r]);
            vS[b * 33 + 16 + ln] = __builtin_amdgcn_exp2f(a1[r] - mD[r]);
        }
        // LDS is in-order within a wave: only block compiler reordering (emits nothing)
        __builtin_amdgcn_wave_barrier();

        // gather all A operands (8x ds_load_2addr_b32), then 4 independent WMMA chains
        // A 16x4 layout: VGPR p, lane -> (M = ln, K = 4*kk + 2*hi + p)
        v2f aops[8];
        #pragma unroll
        for (int kk = 0; kk < 8; ++kk) {
            aops[kk].x = vS[ln * 33 + 4 * kk + 2 * hi];
            aops[kk].y = vS[ln * 33 + 4 * kk + 2 * hi + 1];
        }
        v8f c0e = {}, c0o = {}, c1e = {}, c1o = {};
        #pragma unroll
        for (int kk = 0; kk < 8; kk += 2) {
            c0e = __builtin_amdgcn_wmma_f32_16x16x4_f32(false, aops[kk],     false, Bk[0][kk],
                                                        (short)0, c0e, false, false);
            c1e = __builtin_amdgcn_wmma_f32_16x16x4_f32(false, aops[kk],     false, Bk[1][kk],
                                                        (short)0, c1e, false, false);
            c0o = __builtin_amdgcn_wmma_f32_16x16x4_f32(false, aops[kk + 1], false, Bk[0][kk + 1],
                                                        (short)0, c0o, false, false);
            c1o = __builtin_amdgcn_wmma_f32_16x16x4_f32(false, aops[kk + 1], false, Bk[1][kk + 1],
                                                        (short)0, c1o, false, false);
        }
        __builtin_amdgcn_wave_barrier();   // vS reads precede next iteration's writes

        const v8f c0 = c0e + c0o;   // packed f32 adds
        const v8f c1 = c1e + c1o;

        // beta' = feat*L2E + m2 + log2(w + eps); branchless mask blend:
        // a += mk * (n - a)  (exact for mk in {0,1})
        #pragma unroll
        for (int r = 0; r < 8; ++r) {
            const float n0 = fmaf(f0[r], L2E, mD[r]) + __builtin_amdgcn_logf(c0[r] + 1e-37f);
            const float n1 = fmaf(f1[r], L2E, mD[r]) + __builtin_amdgcn_logf(c1[r] + 1e-37f);
            a0[r] = fmaf(mk[r], n0 - a0[r], a0[r]);
            a1[r] = fmaf(mk[r], n1 - a1[r], a1[r]);
        }
    }

    // ---- out[b] = ln2 * log2sumexp2_j(beta[b,j] + tEOS_j) ----
    float res[8];
    #pragma unroll
    for (int r = 0; r < 8; ++r) {
        const float e0 = a0[r] + tEOS0;
        const float e1 = a1[r] + tEOS1;
        float mm = fmaxf(e0, e1);
        mm = fmaxf(mm, __shfl_xor(mm, 1, 32));
        mm = fmaxf(mm, __shfl_xor(mm, 2, 32));
        mm = fmaxf(mm, __shfl_xor(mm, 4, 32));
        mm = fmaxf(mm, __shfl_xor(mm, 8, 32));
        float s = __builtin_amdgcn_exp2f(e0 - mm) + __builtin_amdgcn_exp2f(e1 - mm);
        s += __shfl_xor(s, 1, 32);
        s += __shfl_xor(s, 2, 32);
        s += __shfl_xor(s, 4, 32);
        s += __shfl_xor(s, 8, 32);
        res[r] = (mm + __builtin_amdgcn_logf(s)) * LN2;
    }
    #pragma unroll
    for (int r = 0; r < 8; ++r) {
        if (hi == 0 && ln == r) out[b0 + r]     = res[r];
        if (hi == 1 && ln == r) out[b0 + 8 + r] = res[r];
    }
}

extern "C" void kernel_launch(void* const* d_in, const int* in_sizes, int n_in,
                              void* d_out, int out_size, void* d_ws, size_t ws_size,
                              hipStream_t stream) {
    const float* feats = (const float*)d_in[0];
    const float* mask  = (const float*)d_in[1];
    const float* trans = (const float*)d_in[2];
    float* out = (float*)d_out;

    const int batch   = out_size;                 // 2048
    const int seq_len = in_sizes[1] / batch;      // mask is (seq_len, batch) -> 512

    const int nblocks = (batch + 15) / 16;        // one wave32 per 16-batch tile
    hipLaunchKernelGGL(crf_fwd_wmma, dim3(nblocks), dim3(32), 0, stream,
                       feats, mask, trans, out, seq_len, batch);
}